// RepulsionLoss_46557445488918
// MI455X (gfx1250) — compile-verified
//
#include <hip/hip_runtime.h>

// CDNA5 / gfx1250 wave32 WMMA implementation of the repulsion loss.
//
// d2[n,m] = |p_n|^2 + |p_m|^2 - 2 p_n.p_m  folded into one V_WMMA_F32_16X16X4_F32:
//   A row n  = [x_n, y_n, z_n, 1]
//   B col m  = [-2x_m, -2y_m, -2z_m, |p_m|^2]
//   s[n,m]   = -2 p_n.p_m + |p_m|^2   (== d2 - |p_n|^2, row-constant shift
//                                       -> preserves per-row top-k ordering)

typedef __attribute__((ext_vector_type(2))) float v2f;
typedef __attribute__((ext_vector_type(8))) float v8f;

#define NPTS   4096
#define NTILES (NPTS / 16)   // 256 column tiles per row sweep
#define KNN    4
#define HRAD   0.03f
#define BIGF   3.0e38f

// Branchless sorted-insert: keep the 5 smallest values in ascending order.
__device__ __forceinline__ void insert5(float (&s)[5], float v) {
    float t = v;
#pragma unroll
    for (int i = 0; i < 5; ++i) {
        float lo = fminf(s[i], t);
        float hi = fmaxf(s[i], t);
        s[i] = lo;
        t = hi;
    }
}

__global__ __launch_bounds__(32) void repulsion_main_kernel(
    const float* __restrict__ pcs,   // [B][NPTS][3]
    float* __restrict__ partials)    // [B * NTILES]
{
    const int blk     = blockIdx.x;          // b * NTILES + rowTile
    const int b       = blk >> 8;            // NTILES == 256
    const int rowTile = blk & (NTILES - 1);
    const int lane    = threadIdx.x;         // 0..31 (wave32)
    const int col     = lane & 15;
    const int half    = lane >> 4;           // 0: K=0,1 | 1: K=2,3 (A/B VGPR layout)

    const float* base = pcs + (size_t)b * NPTS * 3;

    // ---- A tile: rows n0..n0+15, [x y z 1], 16x4 f32 layout (2 VGPRs) ----
    const int n0 = rowTile * 16;
    {
        // nothing; A built below
    }
    const int an = n0 + col;
    const float axv = base[an * 3 + 0];
    const float ayv = base[an * 3 + 1];
    const float azv = base[an * 3 + 2];
    v2f A;
    A.x = half ? azv  : axv;   // K=2 : K=0
    A.y = half ? 1.0f : ayv;   // K=3 : K=1

    // ---- per-(row-slot) top-5 score lists, ascending ----
    float top5[8][5];
#pragma unroll
    for (int r = 0; r < 8; ++r)
#pragma unroll
        for (int j = 0; j < 5; ++j) top5[r][j] = BIGF;

    // ---- sweep all 256 column tiles: one f32 WMMA per tile ----
    for (int t = 0; t < NTILES; ++t) {
        const int m  = t * 16 + col;
        const float qx = base[m * 3 + 0];
        const float qy = base[m * 3 + 1];
        const float qz = base[m * 3 + 2];
        const float qsq = qx * qx + qy * qy + qz * qz;

        v2f Bv;
        Bv.x = half ? (-2.0f * qz) : (-2.0f * qx);  // K=2 : K=0
        Bv.y = half ? qsq          : (-2.0f * qy);  // K=3 : K=1

        v8f c = {};
        // D = A x B + 0  -> 16x16 scores s[n,m] = -2 p_n.p_m + |p_m|^2
        c = __builtin_amdgcn_wmma_f32_16x16x4_f32(
                /*neg_a=*/false, A, /*neg_b=*/false, Bv,
                /*c_mod=*/(short)0, c, /*reuse_a=*/false, /*reuse_b=*/false);

        // C layout: VGPR r, lanes 0-15 -> row M=r, N=lane; lanes 16-31 -> M=r+8.
#pragma unroll
        for (int r = 0; r < 8; ++r) insert5(top5[r], c[r]);
    }

    // ---- merge 16 per-lane top-5 lists per row via LDS ----
    __shared__ float cand[16 * 16 * 5];   // [row][col][5]
    __shared__ float rowContrib[16];

#pragma unroll
    for (int r = 0; r < 8; ++r) {
        const int row = r + half * 8;
#pragma unroll
        for (int j = 0; j < 5; ++j)
            cand[(row * 16 + col) * 5 + j] = top5[r][j];
    }
    __syncthreads();

    if (lane < 16) {
        const int row = lane;
        float best[5] = {BIGF, BIGF, BIGF, BIGF, BIGF};
        for (int cc = 0; cc < 16; ++cc) {
#pragma unroll
            for (int j = 0; j < 5; ++j)
                insert5(best, cand[(row * 16 + cc) * 5 + j]);
        }

        // recover d2 = s + |p_n|^2 ; best[0] is self (d2==0) -> dropped
        const int nn = n0 + row;
        const float px = base[nn * 3 + 0];
        const float py = base[nn * 3 + 1];
        const float pz = base[nn * 3 + 2];
        const float sqn = px * px + py * py + pz * pz;

        float contrib = 0.0f;
#pragma unroll
        for (int k = 1; k <= KNN; ++k) {
            const float d2   = fmaxf(best[k] + sqn, 0.0f);
            const float dist = __builtin_sqrtf(d2 + 1e-12f);
            const float w    = __expf(-d2 * (1.0f / (HRAD * HRAD)));
            contrib += fmaxf(HRAD - dist, 0.0f) * w;
        }
        rowContrib[row] = contrib;
    }
    __syncthreads();

    if (lane == 0) {
        float s = 0.0f;
#pragma unroll
        for (int i = 0; i < 16; ++i) s += rowContrib[i];
        partials[blk] = s;
    }
}

// Deterministic fixed-order final reduction (no float atomics).
__global__ __launch_bounds__(256) void repulsion_reduce_kernel(
    const float* __restrict__ partials, float* __restrict__ out,
    int nPartials, float scale)
{
    __shared__ float buf[256];
    const int tid = threadIdx.x;
    float s = 0.0f;
    for (int i = tid; i < nPartials; i += 256) s += partials[i];
    buf[tid] = s;
    __syncthreads();
#pragma unroll
    for (int off = 128; off > 0; off >>= 1) {
        if (tid < off) buf[tid] += buf[tid + off];
        __syncthreads();
    }
    if (tid == 0) out[0] = buf[0] * scale;
}

extern "C" void kernel_launch(void* const* d_in, const int* in_sizes, int n_in,
                              void* d_out, int out_size, void* d_ws, size_t ws_size,
                              hipStream_t stream) {
    (void)n_in; (void)out_size; (void)ws_size;
    const float* pcs = (const float*)d_in[0];
    float* out       = (float*)d_out;
    float* partials  = (float*)d_ws;              // B*NTILES floats (4 KB at B=4)

    const int B       = in_sizes[0] / (NPTS * 3); // 4 for the reference setup
    const int nBlocks = B * NTILES;               // 1024
    const float scale = 1.0f / (float)(B * NPTS * KNN);

    repulsion_main_kernel<<<dim3(nBlocks), dim3(32), 0, stream>>>(pcs, partials);
    repulsion_reduce_kernel<<<dim3(1), dim3(256), 0, stream>>>(partials, out, nBlocks, scale);
}